// slotGAT_1700807049276
// MI455X (gfx1250) — compile-verified
//
#include <hip/hip_runtime.h>
#include <hip/hip_bf16.h>

typedef __attribute__((ext_vector_type(2))) float v2f;
typedef __attribute__((ext_vector_type(8))) float v8f;

// ---------------------------------------------------------------- constants
#define NNODES 20000
#define NEDGES 200000
#define NPAIRS 10000
#define TSLOT 3
#define NHID 64          // NH
#define DCAT 768         // T*NH*4 emb concat width

#define WMMA_F32(a, b, c) \
  __builtin_amdgcn_wmma_f32_16x16x4_f32(false, (a), false, (b), (short)0, (c), false, false)

// ---------------------------------------------------------------- helpers
__device__ __forceinline__ unsigned f2ord(float f) {
  unsigned u = __float_as_uint(f);
  return (u & 0x80000000u) ? ~u : (u | 0x80000000u);
}
__device__ __forceinline__ float ord2f(unsigned u) {
  return __uint_as_float((u & 0x80000000u) ? (u & 0x7FFFFFFFu) : ~u);
}

__global__ void k_fill_u32(unsigned* __restrict__ p, unsigned v, long n) {
  long i = (long)blockIdx.x * blockDim.x + threadIdx.x;
  if (i < n) p[i] = v;
}

// ---------------------------------------------------------------- input proj (WMMA)
// h[nodeBase+n, t*64+c] = feat[n,:] @ Wm[:,c] + b[c]   (other slots stay 0)
// One wave computes a 16-row x 64-col tile. grid.x = cnt/16 (cnt % 16 == 0).
__global__ void k_inproj_wmma(const float* __restrict__ F, const float* __restrict__ Wm,
                              const float* __restrict__ b, float* __restrict__ h,
                              int K, int nodeBase, int t) {
  const int tm = blockIdx.x;
  const int lane = threadIdx.x;
  const int hi = lane >> 4, lo = lane & 15;
  const int m = tm * 16 + lo;                 // A row this lane supplies
  v8f acc0 = {0.f,0.f,0.f,0.f,0.f,0.f,0.f,0.f};
  v8f acc1 = acc0, acc2 = acc0, acc3 = acc0;
#pragma unroll 2
  for (int k = 0; k < K; k += 4) {
    int c0 = k + 2 * hi;                      // K indices {c0, c0+1} for this half-wave
    v2f a; a.x = F[(long)m * K + c0]; a.y = F[(long)m * K + c0 + 1];
    const float* bp0 = Wm + (long)c0 * 64 + lo;
    const float* bp1 = bp0 + 64;
    v2f b0, b1, b2, b3;
    b0.x = bp0[0];  b0.y = bp1[0];
    b1.x = bp0[16]; b1.y = bp1[16];
    b2.x = bp0[32]; b2.y = bp1[32];
    b3.x = bp0[48]; b3.y = bp1[48];
    acc0 = WMMA_F32(a, b0, acc0);
    acc1 = WMMA_F32(a, b1, acc1);
    acc2 = WMMA_F32(a, b2, acc2);
    acc3 = WMMA_F32(a, b3, acc3);
  }
  v8f accs[4] = {acc0, acc1, acc2, acc3};
#pragma unroll
  for (int j = 0; j < 4; ++j) {
    int col = j * 16 + lo;
    float bias = b[col];
#pragma unroll
    for (int i = 0; i < 8; ++i) {
      int n = tm * 16 + i + 8 * hi;           // D layout: VGPR i -> row i + 8*hi
      h[(long)(nodeBase + n) * (TSLOT * NHID) + t * NHID + col] = accs[j][i] + bias;
    }
  }
}

// ---------------------------------------------------------------- WMMA slot GEMM
// feat[n, hh*192 + t*64 + oo] = sum_c X(n,t,c) * W[t, c, hh*64+oo]
// X(n,t,c) = Xin[n*(Hprev*192) + (c/64)*192 + t*64 + (c%64)]  (head permute folded in)
// One wave computes a 16-row x 64-col tile (one full head block).
// grid = (N/16, H, 3), block = 32. All dims exact for this problem.
__global__ void k_proj_wmma(const float* __restrict__ X, const float* __restrict__ W,
                            float* __restrict__ F, int Hprev, int H) {
  const int C  = Hprev * NHID;
  const int HO = H * NHID;
  const int tm = blockIdx.x, tn = blockIdx.y, t = blockIdx.z;
  const int lane = threadIdx.x;
  const int hi = lane >> 4, lo = lane & 15;
  const int m = tm * 16 + lo;                 // A row this lane supplies
  const long inStride = (long)Hprev * (TSLOT * NHID);
  const float* Wt = W + (long)t * C * HO;
  const int colBase = tn * 64;                // exactly one 64-wide head block
  v8f acc0 = {0.f,0.f,0.f,0.f,0.f,0.f,0.f,0.f};
  v8f acc1 = acc0, acc2 = acc0, acc3 = acc0;
#pragma unroll 2
  for (int k = 0; k < C; k += 4) {
    int c0 = k + 2 * hi;                      // even, so c0 and c0+1 share the 64-block
    long aidx = (long)m * inStride + (long)(c0 >> 6) * (TSLOT * NHID) + t * NHID + (c0 & 63);
    v2f a; a.x = X[aidx]; a.y = X[aidx + 1];
    const float* bp0 = Wt + (long)c0 * HO + colBase + lo;
    const float* bp1 = bp0 + HO;
    v2f b0, b1, b2, b3;
    b0.x = bp0[0];  b0.y = bp1[0];
    b1.x = bp0[16]; b1.y = bp1[16];
    b2.x = bp0[32]; b2.y = bp1[32];
    b3.x = bp0[48]; b3.y = bp1[48];
    acc0 = WMMA_F32(a, b0, acc0);
    acc1 = WMMA_F32(a, b1, acc1);
    acc2 = WMMA_F32(a, b2, acc2);
    acc3 = WMMA_F32(a, b3, acc3);
  }
  v8f accs[4] = {acc0, acc1, acc2, acc3};
  const int hh = colBase >> 6;                // head index of this 64-col block
#pragma unroll
  for (int j = 0; j < 4; ++j) {
    int oo = j * 16 + lo;                     // offset within the head block
#pragma unroll
    for (int i = 0; i < 8; ++i) {
      int n = tm * 16 + i + 8 * hi;
      F[(long)n * (H * TSLOT * NHID) + (long)hh * (TSLOT * NHID) + t * NHID + oo] = accs[j][i];
    }
  }
}

// ---------------------------------------------------------------- attention pieces
__global__ void k_elr(const float* __restrict__ F, const float* __restrict__ al,
                      const float* __restrict__ ar, float* __restrict__ el,
                      float* __restrict__ er, int H) {
  long idx = (long)blockIdx.x * blockDim.x + threadIdx.x;
  if (idx >= (long)NNODES * H) return;
  int n = (int)(idx / H), hh = (int)(idx % H);
  const float* fr = F + (long)n * H * (TSLOT * NHID) + (long)hh * (TSLOT * NHID);
  float sl = 0.f, sr = 0.f;
  for (int d = 0; d < TSLOT * NHID; ++d) {
    float v = fr[d];
    sl = fmaf(v, al[hh * (TSLOT * NHID) + d], sl);
    sr = fmaf(v, ar[hh * (TSLOT * NHID) + d], sr);
  }
  el[idx] = sl; er[idx] = sr;
}

// ee[r,h] = sum_d (ee_tab[r] @ We)[h*64+d] * ae[h,d]  -- tiny, 5*H threads
__global__ void k_ee(const float* __restrict__ ee_tab, const float* __restrict__ We,
                     const float* __restrict__ ae, float* __restrict__ ee, int H) {
  int idx = threadIdx.x;
  if (idx >= 5 * H) return;
  int r = idx / H, hh = idx % H;
  float s = 0.f;
  for (int d = 0; d < 64; ++d) {
    float v = 0.f;
    for (int k = 0; k < 64; ++k)
      v = fmaf(ee_tab[r * 64 + k], We[(long)k * (H * 64) + hh * 64 + d], v);
    s = fmaf(v, ae[hh * 64 + d], s);
  }
  ee[idx] = s;
}

__global__ void k_logit_max(const float* __restrict__ el, const float* __restrict__ er,
                            const float* __restrict__ ee, const int* __restrict__ src,
                            const int* __restrict__ dst, const int* __restrict__ et,
                            float* __restrict__ logit, unsigned* __restrict__ mx, int H) {
  long idx = (long)blockIdx.x * blockDim.x + threadIdx.x;
  if (idx >= (long)NEDGES * H) return;
  int e = (int)(idx / H), hh = (int)(idx % H);
  float l = el[src[e] * H + hh] + er[dst[e] * H + hh] + ee[et[e] * H + hh];
  l = (l >= 0.f) ? l : 0.2f * l;           // leaky_relu
  logit[idx] = l;
  atomicMax(&mx[dst[e] * H + hh], f2ord(l));
}

__global__ void k_exp_sum(const float* __restrict__ logit, const int* __restrict__ dst,
                          const unsigned* __restrict__ mx, float* __restrict__ expv,
                          float* __restrict__ den, int H) {
  long idx = (long)blockIdx.x * blockDim.x + threadIdx.x;
  if (idx >= (long)NEDGES * H) return;
  int e = (int)(idx / H), hh = (int)(idx % H);
  float m = ord2f(mx[dst[e] * H + hh]);
  float ex = expf(logit[idx] - m);
  expv[idx] = ex;
  atomicAdd(&den[dst[e] * H + hh], ex);
}

__global__ void k_attn(const float* __restrict__ expv, const float* __restrict__ den,
                       const int* __restrict__ dst, const float* __restrict__ res,
                       float* __restrict__ aout, int H) {
  long idx = (long)blockIdx.x * blockDim.x + threadIdx.x;
  if (idx >= (long)NEDGES * H) return;
  int e = (int)(idx / H), hh = (int)(idx % H);
  float a = expv[idx] / den[dst[e] * H + hh];
  if (res) a = fmaf(0.05f, res[idx], 0.95f * a);
  aout[idx] = a;
}

// layer-2 special: a is [E,1], residual a1 is [E,4]; broadcast + head-mean collapses
// to abar[e] = 0.95*a2[e] + 0.05*mean_h a1[e,h]
__global__ void k_attn2(const float* __restrict__ expv, const float* __restrict__ den,
                        const int* __restrict__ dst, const float* __restrict__ a1,
                        float* __restrict__ abar) {
  long e = (long)blockIdx.x * blockDim.x + threadIdx.x;
  if (e >= NEDGES) return;
  float a = expv[e] / den[dst[e]];
  float rm = 0.25f * (a1[e * 4 + 0] + a1[e * 4 + 1] + a1[e * 4 + 2] + a1[e * 4 + 3]);
  abar[e] = fmaf(0.05f, rm, 0.95f * a);
}

// out[dst, h, d] += feat[src, h, d] * a[e, h]   (block = 192 threads = one edge)
__global__ void k_agg(const float* __restrict__ F, const float* __restrict__ a,
                      const int* __restrict__ src, const int* __restrict__ dst,
                      float* __restrict__ out, int H) {
  int e = blockIdx.x, d = threadIdx.x;
  int s = src[e], t = dst[e];
  for (int hh = 0; hh < H; ++hh) {
    float v = F[(long)s * H * (TSLOT * NHID) + (long)hh * (TSLOT * NHID) + d] *
              a[(long)e * H + hh];
    atomicAdd(&out[(long)t * H * (TSLOT * NHID) + (long)hh * (TSLOT * NHID) + d], v);
  }
}

__global__ void k_elu(float* __restrict__ x, long n) {
  long i = (long)blockIdx.x * blockDim.x + threadIdx.x;
  if (i >= n) return;
  float v = x[i];
  x[i] = (v > 0.f) ? v : (expf(v) - 1.f);
}

// o[n, t*256 + embIdx*64 + c] = mean_h src[n, h, t*64+c]
__global__ void k_mean_copy(const float* __restrict__ src, float* __restrict__ o,
                            int Hh, int embIdx) {
  long idx = (long)blockIdx.x * blockDim.x + threadIdx.x;
  if (idx >= (long)NNODES * (TSLOT * NHID)) return;
  int n = (int)(idx / (TSLOT * NHID)), d = (int)(idx % (TSLOT * NHID));
  int t = d >> 6, c = d & 63;
  float s = 0.f;
  for (int hh = 0; hh < Hh; ++hh)
    s += src[(long)n * Hh * (TSLOT * NHID) + (long)hh * (TSLOT * NHID) + d];
  o[(long)n * DCAT + t * 256 + embIdx * 64 + c] = s / (float)Hh;
}

// per-slot l2 normalize the just-written 64-wide segment in place
__global__ void k_l2slot(float* __restrict__ o, int embIdx) {
  long idx = (long)blockIdx.x * blockDim.x + threadIdx.x;
  if (idx >= (long)NNODES * TSLOT) return;
  int n = (int)(idx / TSLOT), t = (int)(idx % TSLOT);
  float* p = o + (long)n * DCAT + t * 256 + embIdx * 64;
  float ss = 0.f;
  for (int c = 0; c < 64; ++c) ss = fmaf(p[c], p[c], ss);
  float inv = 1.f / fmaxf(sqrtf(ss), 1e-12f);
  for (int c = 0; c < 64; ++c) p[c] *= inv;
}

// ---------------------------------------------------------------- DistMult scoring
// score[i] = sigmoid( sum_j (sum_k o[left,k] * W[mid][k,j]) * o[right,j] )
__global__ void k_distmult(const float* __restrict__ o, const float* __restrict__ dmW,
                           const int* __restrict__ left, const int* __restrict__ right,
                           const int* __restrict__ mid, float* __restrict__ outp) {
  __shared__ float le[DCAT];
  __shared__ float red[256];
  int i = blockIdx.x, tid = threadIdx.x;
  const float* lrow = o + (long)left[i] * DCAT;
  const float* rrow = o + (long)right[i] * DCAT;
  for (int k = tid; k < DCAT; k += 256) le[k] = lrow[k];
  __syncthreads();
  const float* Wr = dmW + (long)mid[i] * DCAT * DCAT;
  float acc = 0.f;
  for (int j = tid; j < DCAT; j += 256) {
    float t = 0.f;
    for (int k = 0; k < DCAT; ++k) t = fmaf(le[k], Wr[(long)k * DCAT + j], t);
    acc = fmaf(t, rrow[j], acc);
  }
  red[tid] = acc;
  __syncthreads();
  for (int s = 128; s > 0; s >>= 1) {
    if (tid < s) red[tid] += red[tid + s];
    __syncthreads();
  }
  if (tid == 0) outp[i] = 1.f / (1.f + expf(-red[0]));
}

// ---------------------------------------------------------------- host
extern "C" void kernel_launch(void* const* d_in, const int* in_sizes, int n_in,
                              void* d_out, int out_size, void* d_ws, size_t ws_size,
                              hipStream_t stream) {
  const float* feat[3] = {(const float*)d_in[0], (const float*)d_in[3], (const float*)d_in[6]};
  const float* w_in[3] = {(const float*)d_in[1], (const float*)d_in[4], (const float*)d_in[7]};
  const float* b_in[3] = {(const float*)d_in[2], (const float*)d_in[5], (const float*)d_in[8]};
  const float* lW[3]   = {(const float*)d_in[9],  (const float*)d_in[15], (const float*)d_in[21]};
  const float* lal[3]  = {(const float*)d_in[10], (const float*)d_in[16], (const float*)d_in[22]};
  const float* lar[3]  = {(const float*)d_in[11], (const float*)d_in[17], (const float*)d_in[23]};
  const float* leeT[3] = {(const float*)d_in[12], (const float*)d_in[18], (const float*)d_in[24]};
  const float* lwe[3]  = {(const float*)d_in[13], (const float*)d_in[19], (const float*)d_in[25]};
  const float* lae[3]  = {(const float*)d_in[14], (const float*)d_in[20], (const float*)d_in[26]};
  const float* dmW = (const float*)d_in[27];
  const int* e_feat = (const int*)d_in[28];
  const int* srcI   = (const int*)d_in[29];
  const int* dstI   = (const int*)d_in[30];
  const int* leftI  = (const int*)d_in[31];
  const int* rightI = (const int*)d_in[32];
  const int* midI   = (const int*)d_in[33];
  float* scores = (float*)d_out;

  // workspace carve (floats)
  float* ws = (float*)d_ws;
  long off = 0;
  auto alloc = [&](long n) { float* p = ws + off; off += n; return p; };
  float* h     = alloc((long)NNODES * 192);       // slot-padded input projection
  float* featb = alloc((long)NNODES * 4 * 192);   // per-layer projected features
  float* outA  = alloc((long)NNODES * 4 * 192);   // layer0 output
  float* outB  = alloc((long)NNODES * 4 * 192);   // layer1 output
  float* out2m = alloc((long)NNODES * 192);       // layer2 head-mean output
  float* obuf  = alloc((long)NNODES * DCAT);      // final emb concat [N,768]
  float* el    = alloc((long)NNODES * 4);
  float* er    = alloc((long)NNODES * 4);
  float* den   = alloc((long)NNODES * 4);
  unsigned* mx = (unsigned*)alloc((long)NNODES * 4);
  float* logit = alloc((long)NEDGES * 4);
  float* expv  = alloc((long)NEDGES * 4);
  float* a0    = alloc((long)NEDGES * 4);
  float* a1    = alloc((long)NEDGES * 4);
  float* abar  = alloc((long)NEDGES);
  float* eeTab = alloc(32);

  auto zero = [&](void* p, long n32) {
    k_fill_u32<<<(unsigned)((n32 + 255) / 256), 256, 0, stream>>>((unsigned*)p, 0u, n32);
  };
  const int counts[3] = {8000, 6000, 6000};
  const int indims[3] = {256, 128, 64};
  const int bases[3]  = {0, 8000, 14000};

  // ---- input projection into slot-padded h (WMMA), emb0 into o
  zero(h, (long)NNODES * 192);
  for (int t = 0; t < 3; ++t) {
    k_inproj_wmma<<<counts[t] / 16, 32, 0, stream>>>(
        feat[t], w_in[t], b_in[t], h, indims[t], bases[t], t);
  }
  long nND = (long)NNODES * 192;
  k_mean_copy<<<(unsigned)((nND + 255) / 256), 256, 0, stream>>>(h, obuf, 1, 0);
  k_l2slot<<<(NNODES * TSLOT + 255) / 256, 256, 0, stream>>>(obuf, 0);

  // ---- three GAT layers
  const int   Hs[3]     = {4, 4, 1};
  const int   Hprevs[3] = {1, 4, 4};
  const float* xin[3]   = {h, outA, outB};
  float*      outs[3]   = {outA, outB, out2m};
  float*      aouts[3]  = {a0, a1, abar};

  for (int l = 0; l < 3; ++l) {
    const int H = Hs[l], Hp = Hprevs[l];
    // slot-wise projection (WMMA f32): one wave per 16x64 tile, grid (N/16, H, 3)
    dim3 g(NNODES / 16, H, 3);
    k_proj_wmma<<<g, 32, 0, stream>>>(xin[l], lW[l], featb, Hp, H);
    // node attention terms
    long nNH = (long)NNODES * H;
    k_elr<<<(unsigned)((nNH + 255) / 256), 256, 0, stream>>>(featb, lal[l], lar[l], el, er, H);
    // per-etype edge term
    k_ee<<<1, 32, 0, stream>>>(leeT[l], lwe[l], lae[l], eeTab, H);
    // edge softmax
    zero(mx, nNH);
    zero(den, nNH);
    long nEH = (long)NEDGES * H;
    k_logit_max<<<(unsigned)((nEH + 255) / 256), 256, 0, stream>>>(
        el, er, eeTab, srcI, dstI, e_feat, logit, mx, H);
    k_exp_sum<<<(unsigned)((nEH + 255) / 256), 256, 0, stream>>>(logit, dstI, mx, expv, den, H);
    if (l < 2) {
      k_attn<<<(unsigned)((nEH + 255) / 256), 256, 0, stream>>>(
          expv, den, dstI, (l == 0) ? (const float*)nullptr : a0, aouts[l], H);
    } else {
      k_attn2<<<(NEDGES + 255) / 256, 256, 0, stream>>>(expv, den, dstI, a1, abar);
    }
    // aggregate
    long outN = (long)NNODES * ((l == 2) ? 1 : H) * 192;
    zero(outs[l], outN);
    k_agg<<<NEDGES, 192, 0, stream>>>(featb, aouts[l], srcI, dstI, outs[l], (l == 2) ? 1 : H);
    if (l < 2) k_elu<<<(unsigned)((outN + 255) / 256), 256, 0, stream>>>(outs[l], outN);
    // emb_{l+1}
    k_mean_copy<<<(unsigned)((nND + 255) / 256), 256, 0, stream>>>(
        outs[l], obuf, (l == 2) ? 1 : H, l + 1);
    k_l2slot<<<(NNODES * TSLOT + 255) / 256, 256, 0, stream>>>(obuf, l + 1);
  }

  // ---- DistMult scoring
  k_distmult<<<NPAIRS, 256, 0, stream>>>(obuf, dmW, leftI, rightI, midI, scores);
}